// MoFE_82111184765072
// MI455X (gfx1250) — compile-verified
//
#include <hip/hip_runtime.h>
#include <math.h>

// ---------------- problem constants ----------------
#define Bn  8
#define Cn  64
#define Hn  128
#define Wn  128
#define HWn (Hn*Wn)       // 16384
#define H2  64
#define W2  64
#define HW2 (H2*W2)       // 4096
#define NEe 4
#define TPB 256

typedef float v2f __attribute__((ext_vector_type(2)));
typedef float v8f __attribute__((ext_vector_type(8)));

// ---------------- gate coefficients (all 3 gates share pooled stats of x) ----
__global__ void k_gate(const float* __restrict__ x,
                       const float* fc0w0, const float* fc0b0, const float* fc1w0, const float* fc1b0,
                       const float* fc0w1, const float* fc0b1, const float* fc1w1, const float* fc1b1,
                       const float* fc0w2, const float* fc0b2, const float* fc1w2, const float* fc1b2,
                       float* __restrict__ coef) {
  __shared__ float pooled[Cn];
  __shared__ float red[TPB];
  const int b = blockIdx.x, t = threadIdx.x;
  const float* xp = x + (size_t)b*Cn*HWn;
  for (int c = 0; c < Cn; ++c) {
    float mx = -3.0e38f, s = 0.f;
    for (int i = t; i < HWn; i += TPB) { float v = xp[(size_t)c*HWn + i]; mx = fmaxf(mx, v); s += v; }
    red[t] = mx; __syncthreads();
    for (int st = TPB/2; st > 0; st >>= 1) { if (t < st) red[t] = fmaxf(red[t], red[t+st]); __syncthreads(); }
    float cmax = red[0]; __syncthreads();
    red[t] = s; __syncthreads();
    for (int st = TPB/2; st > 0; st >>= 1) { if (t < st) red[t] += red[t+st]; __syncthreads(); }
    if (t == 0) pooled[c] = cmax + red[0]*(1.0f/HWn);
    __syncthreads();
  }
  if (t == 0) {
    const float* W0[3] = {fc0w0, fc0w1, fc0w2};
    const float* B0[3] = {fc0b0, fc0b1, fc0b2};
    const float* W1[3] = {fc1w0, fc1w1, fc1w2};
    const float* B1[3] = {fc1b0, fc1b1, fc1b2};
    for (int g = 0; g < 3; ++g) {
      float h[NEe], nz[NEe];
      for (int e = 0; e < NEe; ++e) {
        float d0 = B0[g][e], d1 = B1[g][e];
        for (int c = 0; c < Cn; ++c) { d0 += pooled[c]*W0[g][e*Cn+c]; d1 += pooled[c]*W1[g][e*Cn+c]; }
        h[e]  = d1 >= 0.f ? d1 : 0.2f*d1;                 // leaky 0.2
        nz[e] = d0 > 20.f ? d0 : log1pf(expf(d0));        // softplus
      }
      float sm = 0.f; for (int e = 0; e < NEe; ++e) sm += nz[e]; sm *= 0.25f;
      float var = 0.f; for (int e = 0; e < NEe; ++e) { float d = nz[e]-sm; var += d*d; }
      float ns = sqrtf(var*(1.f/3.f)) + 1e-6f;            // ddof=1
      float sc[NEe]; for (int e = 0; e < NEe; ++e) sc[e] = h[e] + (nz[e]-sm)/ns;
      int mi = 0; float mv = sc[0];
      for (int e = 1; e < NEe; ++e) if (sc[e] <= mv) { mv = sc[e]; mi = e; }  // top_k: last tied min loses
      float hm = -3.0e38f;
      for (int e = 0; e < NEe; ++e) if (e != mi) hm = fmaxf(hm, h[e]);
      float se = 0.f, cf[NEe];
      for (int e = 0; e < NEe; ++e) { cf[e] = (e == mi) ? 0.f : expf(h[e]-hm); se += cf[e]; }
      float inv = 1.f/se;
      for (int e = 0; e < NEe; ++e) coef[(g*Bn + b)*NEe + e] = cf[e]*inv;
    }
  }
}

// ---------------- zero a big buffer ----------------
__global__ void k_zero(float* __restrict__ o) {
  o[(size_t)blockIdx.x*TPB + threadIdx.x] = 0.f;
}

// ---------------- depthwise 3x3 pad1 ----------------
__global__ void k_dwconv(const float* __restrict__ in, const float* __restrict__ w,
                         const float* __restrict__ bias, float* __restrict__ out) {
  size_t n = (size_t)blockIdx.x*TPB + threadIdx.x;      // B*C*HW
  int p = (int)(n % HWn); int c = (int)((n / HWn) % Cn); int b = (int)(n / ((size_t)HWn*Cn));
  int y = p / Wn, xx = p % Wn;
  const float* ip = in + ((size_t)b*Cn + c)*HWn;
  const float* wp = w + c*9;
  float s = bias[c];
  #pragma unroll
  for (int ky = 0; ky < 3; ++ky) {
    int yy = y + ky - 1; if ((unsigned)yy >= (unsigned)Hn) continue;
    #pragma unroll
    for (int kx = 0; kx < 3; ++kx) {
      int xc = xx + kx - 1; if ((unsigned)xc >= (unsigned)Wn) continue;
      s += ip[yy*Wn + xc]*wp[ky*3 + kx];
    }
  }
  out[n] = s;
}

// ---------------- depthwise 3x3 pad1, scaled accumulate (expert c2) ---------
__global__ void k_dwacc(const float* __restrict__ in, const float* __restrict__ w,
                        const float* __restrict__ bias, const float* __restrict__ coef,
                        float* __restrict__ out) {
  size_t n = (size_t)blockIdx.x*TPB + threadIdx.x;
  int p = (int)(n % HWn); int c = (int)((n / HWn) % Cn); int b = (int)(n / ((size_t)HWn*Cn));
  int y = p / Wn, xx = p % Wn;
  const float* ip = in + ((size_t)b*Cn + c)*HWn;
  const float* wp = w + c*9;
  float s = bias[c];
  #pragma unroll
  for (int ky = 0; ky < 3; ++ky) {
    int yy = y + ky - 1; if ((unsigned)yy >= (unsigned)Hn) continue;
    #pragma unroll
    for (int kx = 0; kx < 3; ++kx) {
      int xc = xx + kx - 1; if ((unsigned)xc >= (unsigned)Wn) continue;
      s += ip[yy*Wn + xc]*wp[ky*3 + kx];
    }
  }
  out[n] += coef[b*NEe]*s;
}

// ---------------- channel pooling (avg=0, max=1, std=2) ----------------
__global__ void k_pool(const float* __restrict__ t1, int m1, int m2,
                       float* __restrict__ pm1, float* __restrict__ pm2) {
  size_t n = (size_t)blockIdx.x*TPB + threadIdx.x;      // B*HW
  int p = (int)(n % HWn); int b = (int)(n / HWn);
  const float* ip = t1 + (size_t)b*Cn*HWn + p;
  float s = 0.f, sq = 0.f, mx = -3.0e38f;
  for (int c = 0; c < Cn; ++c) { float v = ip[(size_t)c*HWn]; s += v; sq += v*v; mx = fmaxf(mx, v); }
  float mean = s*(1.f/Cn), msq = sq*(1.f/Cn);
  float stdv = sqrtf(fmaxf(msq - mean*mean, 1e-6f));
  pm1[n] = (m1 == 0) ? mean : ((m1 == 1) ? mx : stdv);
  pm2[n] = (m2 == 0) ? mean : ((m2 == 1) ? mx : stdv);
}

// ---------------- 1-ch 3x3 pad1 stride2 (128 -> 64) ----------------
__global__ void k_ds(const float* __restrict__ in, const float* __restrict__ w9,
                     const float* __restrict__ b1, float* __restrict__ out) {
  size_t n = (size_t)blockIdx.x*TPB + threadIdx.x;      // B*HW2
  int p = (int)(n % HW2); int b = (int)(n / HW2);
  int y = p / W2, xx = p % W2;
  const float* ip = in + (size_t)b*HWn;
  float s = b1[0];
  #pragma unroll
  for (int ky = 0; ky < 3; ++ky) {
    int yy = 2*y + ky - 1; if ((unsigned)yy >= (unsigned)Hn) continue;
    #pragma unroll
    for (int kx = 0; kx < 3; ++kx) {
      int xc = 2*xx + kx - 1; if ((unsigned)xc >= (unsigned)Wn) continue;
      s += ip[yy*Wn + xc]*w9[ky*3 + kx];
    }
  }
  out[n] = s;
}

// ---------------- 1-ch 3x3 pad1 on 64x64, act: 0 none, 1 leaky(.01), 2 tanh -
__global__ void k_c1(const float* __restrict__ in, const float* __restrict__ w9,
                     const float* __restrict__ b1, float* __restrict__ out, int act) {
  size_t n = (size_t)blockIdx.x*TPB + threadIdx.x;      // B*HW2
  int p = (int)(n % HW2); int b = (int)(n / HW2);
  int y = p / W2, xx = p % W2;
  const float* ip = in + (size_t)b*HW2;
  float s = b1[0];
  #pragma unroll
  for (int ky = 0; ky < 3; ++ky) {
    int yy = y + ky - 1; if ((unsigned)yy >= (unsigned)H2) continue;
    #pragma unroll
    for (int kx = 0; kx < 3; ++kx) {
      int xc = xx + kx - 1; if ((unsigned)xc >= (unsigned)W2) continue;
      s += ip[yy*W2 + xc]*w9[ky*3 + kx];
    }
  }
  if (act == 1)      s = s >= 0.f ? s : 0.01f*s;
  else if (act == 2) s = tanhf(s);
  out[n] = s;
}

// ---------------- cross-mix: ch0=f1*w2+b2, ch1=f2*w1+b1 ----------------
__global__ void k_fusemix(const float* __restrict__ f1, const float* __restrict__ f2,
                          const float* __restrict__ w1, const float* __restrict__ b1v,
                          const float* __restrict__ w2, const float* __restrict__ b2v,
                          float* __restrict__ o0, float* __restrict__ o1) {
  size_t n = (size_t)blockIdx.x*TPB + threadIdx.x;
  o0[n] = f1[n]*w2[n] + b2v[n];
  o1[n] = f2[n]*w1[n] + b1v[n];
}

// ---------------- 2-ch 3x3 pad1 fuse conv on 64x64 ----------------
__global__ void k_c2ch(const float* __restrict__ i0, const float* __restrict__ i1,
                       const float* __restrict__ w18, const float* __restrict__ b1,
                       float* __restrict__ out) {
  size_t n = (size_t)blockIdx.x*TPB + threadIdx.x;
  int p = (int)(n % HW2); int b = (int)(n / HW2);
  int y = p / W2, xx = p % W2;
  const float* a0 = i0 + (size_t)b*HW2;
  const float* a1 = i1 + (size_t)b*HW2;
  float s = b1[0];
  #pragma unroll
  for (int ky = 0; ky < 3; ++ky) {
    int yy = y + ky - 1; if ((unsigned)yy >= (unsigned)H2) continue;
    #pragma unroll
    for (int kx = 0; kx < 3; ++kx) {
      int xc = xx + kx - 1; if ((unsigned)xc >= (unsigned)W2) continue;
      s += a0[yy*W2 + xc]*w18[ky*3 + kx] + a1[yy*W2 + xc]*w18[9 + ky*3 + kx];
    }
  }
  out[n] = s;
}

// ---------------- bilinear x2 (align_corners=False) + sigmoid ---------------
__global__ void k_upsig(const float* __restrict__ e, float* __restrict__ up) {
  size_t n = (size_t)blockIdx.x*TPB + threadIdx.x;      // B*HWn
  int p = (int)(n % HWn); int b = (int)(n / HWn);
  int y = p / Wn, xx = p % Wn;
  float sy = 0.5f*(float)y - 0.25f, sx = 0.5f*(float)xx - 0.25f;
  float fy = floorf(sy), fx = floorf(sx);
  float ty = sy - fy, tx = sx - fx;
  int y0 = min(max((int)fy, 0), H2-1),     x0 = min(max((int)fx, 0), W2-1);
  int y1 = min(max((int)fy + 1, 0), H2-1), x1 = min(max((int)fx + 1, 0), W2-1);
  const float* ep = e + (size_t)b*HW2;
  float v = (1.f-ty)*((1.f-tx)*ep[y0*W2+x0] + tx*ep[y0*W2+x1])
          +      ty *((1.f-tx)*ep[y1*W2+x0] + tx*ep[y1*W2+x1]);
  up[n] = 1.f/(1.f + expf(-v));
}

// ---------------- t2 = relu(t1 * up) ----------------
__global__ void k_gateapply(const float* __restrict__ t1, const float* __restrict__ up,
                            float* __restrict__ t2) {
  size_t n = (size_t)blockIdx.x*TPB + threadIdx.x;      // B*C*HW
  int p = (int)(n % HWn); int b = (int)(n / ((size_t)HWn*Cn));
  float v = t1[n]*up[(size_t)b*HWn + p];
  t2[n] = fmaxf(v, 0.f);
}

// ---------------- final 1x1 conv: per-gate fp32 WMMA GEMM -------------------
// out[b,o,p] (init: = bias[o] +) += sum_k W[o, g*64+k] * G[b,k,p]
// One wave computes a 64(o) x 32(p) tile: 4 o-tiles x 2 p-subtiles =
// 8 v8f accumulators. Each B operand pair is loaded ONCE and reused by all 4
// o-tiles, so G (the 33.5 MB operand) is streamed from memory exactly once;
// the A operand (48 KB weight matrix) is L2-resident.
// V_WMMA_F32_16X16X4_F32 operand layouts:
//   A 16x4: lanes 0-15 = M rows; VGPR0 -> K=2*hi, VGPR1 -> K=2*hi+1
//   B 4x16: lane -> N=lane&15;   VGPR0 -> K=2*hi, VGPR1 -> K=2*hi+1
//   C/D 16x16: lane -> N=lane&15, VGPR v -> M = v + 8*hi
__global__ void k_gemm(const float* __restrict__ G,      // [B,64,HW]
                       const float* __restrict__ Wf,     // [64,192]
                       const float* __restrict__ bias,   // [64]
                       int g, int init, float* __restrict__ out) {
  int wid  = (int)(((size_t)blockIdx.x*blockDim.x + threadIdx.x) >> 5);
  int lane = threadIdx.x & 31;
  int pt = wid & 511;            // 512 p-tiles of 32
  int b  = wid >> 9;             // batch
  int p0 = pt*32;
  int m  = lane & 15;
  int hi = lane >> 4;
  const float* Wbase = Wf + (size_t)m*192 + g*64 + 2*hi;       // + t*16*192 + kk
  const float* Gb    = G + (size_t)b*Cn*HWn + (size_t)(2*hi)*HWn + p0 + m;
  v8f z = {0.f,0.f,0.f,0.f,0.f,0.f,0.f,0.f};
  v8f acc[4][2];
  #pragma unroll
  for (int t = 0; t < 4; ++t) { acc[t][0] = z; acc[t][1] = z; }
  for (int kk = 0; kk < Cn; kk += 4) {
    const float* grow = Gb + (size_t)kk*HWn;
    // prefetch next k-step's G rows into cache (global_prefetch_b8)
    __builtin_prefetch(grow + (size_t)4*HWn, 0, 0);
    __builtin_prefetch(grow + (size_t)5*HWn, 0, 0);
    v2f b0; b0.x = grow[0];  b0.y = grow[HWn];
    v2f b1; b1.x = grow[16]; b1.y = grow[HWn + 16];
    #pragma unroll
    for (int t = 0; t < 4; ++t) {
      const float* wr = Wbase + (size_t)t*16*192 + kk;
      v2f a; a.x = wr[0]; a.y = wr[1];
      acc[t][0] = __builtin_amdgcn_wmma_f32_16x16x4_f32(false, a, false, b0, (short)0, acc[t][0], false, false);
      acc[t][1] = __builtin_amdgcn_wmma_f32_16x16x4_f32(false, a, false, b1, (short)0, acc[t][1], false, false);
    }
  }
  #pragma unroll
  for (int t = 0; t < 4; ++t) {
    #pragma unroll
    for (int j = 0; j < 2; ++j) {
      int p = p0 + j*16 + m;
      #pragma unroll
      for (int v = 0; v < 8; ++v) {
        int o = t*16 + v + 8*hi;
        size_t oi = ((size_t)b*Cn + o)*HWn + p;
        float r = acc[t][j][v];
        if (init) out[oi] = bias[o] + r;
        else      out[oi] += r;
      }
    }
  }
}

// ---------------- host launcher ----------------
// Input flattening order (dict insertion order, recursive):
//   0: x [8,64,128,128]
//   1..12: gates g0..g2: fc0_w, fc0_b, fc1_w, fc1_b
//   13..276: experts (g,e) base = 13 + (g*4+e)*22:
//     +0 c1_w +1 c1_b +2 ds1_w +3 ds1_b +4 ds2_w +5 ds2_b
//     +6..+11 m0_w,m0_b,m1_w,m1_b,m2_w,m2_b
//     +12..+17 b0_w,b0_b,b1_w,b1_b,b2_w,b2_b
//     +18 efuse_w +19 efuse_b +20 c2_w +21 c2_b
//   277: fuse_w [64,192]   278: fuse_b [64]
extern "C" void kernel_launch(void* const* d_in, const int* in_sizes, int n_in,
                              void* d_out, int out_size, void* d_ws, size_t ws_size,
                              hipStream_t stream) {
  auto F = [&](int i) -> const float* { return (const float*)d_in[i]; };
  const float* x = F(0);

  float* ws = (float*)d_ws;
  size_t off = 0;
  float* GOUT = ws + off; off += (size_t)Bn*Cn*HWn;   // per-gate expert mix
  float* T1   = ws + off; off += (size_t)Bn*Cn*HWn;   // depthwise c1 output
  float* T2   = ws + off; off += (size_t)Bn*Cn*HWn;   // emb output
  float* PM1  = ws + off; off += (size_t)Bn*HWn;
  float* PM2  = ws + off; off += (size_t)Bn*HWn;
  float* UP   = ws + off; off += (size_t)Bn*HWn;
  float* F1D  = ws + off; off += (size_t)Bn*HW2;
  float* F2D  = ws + off; off += (size_t)Bn*HW2;
  float* W1b  = ws + off; off += (size_t)Bn*HW2;
  float* B1b  = ws + off; off += (size_t)Bn*HW2;
  float* W2b  = ws + off; off += (size_t)Bn*HW2;
  float* B2b  = ws + off; off += (size_t)Bn*HW2;
  float* TMPA = ws + off; off += (size_t)Bn*HW2;
  float* TMPB = ws + off; off += (size_t)Bn*HW2;
  float* EOUT = ws + off; off += (size_t)Bn*HW2;
  float* COEF = ws + off; off += (size_t)3*Bn*NEe;

  const int NB_BCHW = (Bn*Cn*HWn)/TPB;   // 32768
  const int NB_BHW  = (Bn*HWn)/TPB;      // 512
  const int NB_BHW2 = (Bn*HW2)/TPB;      // 128

  // gate coefficients for all 3 gates (pooled stats of x shared)
  k_gate<<<Bn, TPB, 0, stream>>>(x, F(1),F(2),F(3),F(4), F(5),F(6),F(7),F(8),
                                 F(9),F(10),F(11),F(12), COEF);

  static const int M1[3] = {0, 2, 2};   // avg, std, std
  static const int M2[3] = {1, 1, 0};   // max, max, avg

  for (int g = 0; g < 3; ++g) {
    k_zero<<<NB_BCHW, TPB, 0, stream>>>(GOUT);
    for (int e = 0; e < NEe; ++e) {
      const int base = 13 + (g*NEe + e)*22;
      // depthwise c1
      k_dwconv<<<NB_BCHW, TPB, 0, stream>>>(x, F(base+0), F(base+1), T1);
      // channel pooling (both modes, one pass)
      k_pool<<<NB_BHW, TPB, 0, stream>>>(T1, M1[g], M2[g], PM1, PM2);
      // downsample convs
      k_ds<<<NB_BHW2, TPB, 0, stream>>>(PM1, F(base+2), F(base+3), F1D);
      k_ds<<<NB_BHW2, TPB, 0, stream>>>(PM2, F(base+4), F(base+5), F2D);
      // branches (m-params shared by w1/w2; b-params by b1/b2)
      k_c1<<<NB_BHW2, TPB, 0, stream>>>(F1D,  F(base+6),  F(base+7),  TMPA, 1);
      k_c1<<<NB_BHW2, TPB, 0, stream>>>(TMPA, F(base+8),  F(base+9),  TMPB, 1);
      k_c1<<<NB_BHW2, TPB, 0, stream>>>(TMPB, F(base+10), F(base+11), W1b,  2);
      k_c1<<<NB_BHW2, TPB, 0, stream>>>(F1D,  F(base+12), F(base+13), TMPA, 1);
      k_c1<<<NB_BHW2, TPB, 0, stream>>>(TMPA, F(base+14), F(base+15), TMPB, 1);
      k_c1<<<NB_BHW2, TPB, 0, stream>>>(TMPB, F(base+16), F(base+17), B1b,  0);
      k_c1<<<NB_BHW2, TPB, 0, stream>>>(F2D,  F(base+6),  F(base+7),  TMPA, 1);
      k_c1<<<NB_BHW2, TPB, 0, stream>>>(TMPA, F(base+8),  F(base+9),  TMPB, 1);
      k_c1<<<NB_BHW2, TPB, 0, stream>>>(TMPB, F(base+10), F(base+11), W2b,  2);
      k_c1<<<NB_BHW2, TPB, 0, stream>>>(F2D,  F(base+12), F(base+13), TMPA, 1);
      k_c1<<<NB_BHW2, TPB, 0, stream>>>(TMPA, F(base+14), F(base+15), TMPB, 1);
      k_c1<<<NB_BHW2, TPB, 0, stream>>>(TMPB, F(base+16), F(base+17), B2b,  0);
      // cross-mix + 2ch fuse conv
      k_fusemix<<<NB_BHW2, TPB, 0, stream>>>(F1D, F2D, W1b, B1b, W2b, B2b, TMPA, TMPB);
      k_c2ch<<<NB_BHW2, TPB, 0, stream>>>(TMPA, TMPB, F(base+18), F(base+19), EOUT);
      // upsample+sigmoid, gated relu
      k_upsig<<<NB_BHW, TPB, 0, stream>>>(EOUT, UP);
      k_gateapply<<<NB_BCHW, TPB, 0, stream>>>(T1, UP, T2);
      // depthwise c2, scaled by gate coefficient, accumulate into GOUT
      k_dwacc<<<NB_BCHW, TPB, 0, stream>>>(T2, F(base+20), F(base+21),
                                           COEF + (size_t)g*Bn*NEe + e, GOUT);
    }
    // final 1x1 conv slice for this gate via fp32 WMMA (4096 waves,
    // 64x32 tile per wave, single-pass streaming of GOUT)
    k_gemm<<<512, 256, 0, stream>>>(GOUT, F(277), F(278), g, (g == 0) ? 1 : 0,
                                    (float*)d_out);
  }
}